// Lorenz96_dict_param_7421703487573
// MI455X (gfx1250) — compile-verified
//
#include <hip/hip_runtime.h>

typedef __attribute__((ext_vector_type(2))) float v2f;
typedef __attribute__((ext_vector_type(4))) float v4f;
typedef __attribute__((ext_vector_type(8))) float v8f;

#define XD 40
#define ROWS_PER_WAVE 4
#define WAVES_PER_BLOCK 8
#define ROWS_PER_BLOCK (ROWS_PER_WAVE * WAVES_PER_BLOCK)      // 32
#define ELEMS_PER_WAVE (XD * ROWS_PER_WAVE)                   // 160

// Build the 18 Lorenz96 dictionary features (padded to 20 with zeros) for
// element index e (0..159) within this wave's 4-row LDS tile. Circular shifts
// wrap within each 40-wide row.
__device__ __forceinline__ void compute_feats(const float* __restrict__ tile,
                                              int e, float* __restrict__ f) {
    int row = e / XD;
    int i   = e - row * XD;
    const float* r = tile + row * XD;
    int im2 = i - 2; im2 += (im2 < 0) ? XD : 0;
    int im1 = i - 1; im1 += (im1 < 0) ? XD : 0;
    int ip1 = i + 1; ip1 -= (ip1 >= XD) ? XD : 0;
    int ip2 = i + 2; ip2 -= (ip2 >= XD) ? XD : 0;
    float um2 = r[im2], um1 = r[im1], uc = r[i], up1 = r[ip1], up2 = r[ip2];
    f[0]  = 1.0f;
    f[1]  = um2;       f[2]  = um1;       f[3]  = uc;
    f[4]  = up1;       f[5]  = up2;
    f[6]  = um2 * um2; f[7]  = um1 * um1; f[8]  = uc * uc;
    f[9]  = up1 * up1; f[10] = up2 * up2;
    f[11] = um2 * um1; f[12] = um1 * uc;  f[13] = uc * up1;
    f[14] = up1 * up2;
    f[15] = um2 * uc;  f[16] = um1 * up1; f[17] = uc * up2;
    f[18] = 0.0f;      f[19] = 0.0f;
}

__global__ __launch_bounds__(WAVES_PER_BLOCK * 32)
void l96_feat_dot_kernel(const float* __restrict__ u,
                         const float* __restrict__ coeff,
                         float* __restrict__ out,
                         int total_rows) {
    __shared__ float lds_tile[WAVES_PER_BLOCK][ELEMS_PER_WAVE];

    const int tid  = threadIdx.x;
    const int wave = tid >> 5;
    const int lane = tid & 31;
    const bool hi  = (lane >= 16);   // half-wave carries K=2,3 of each K4 step
    const int n    = lane & 15;      // output column within a 16-chunk

    const long rowBase = (long)blockIdx.x * ROWS_PER_BLOCK + (long)wave * ROWS_PER_WAVE;
    if (rowBase + ROWS_PER_WAVE > (long)total_rows) return;  // wave-uniform

    const float* __restrict__ src = u   + rowBase * XD;
    float* __restrict__       dst = out + rowBase * XD;
    float* __restrict__       lds = lds_tile[wave];

    // ---- Stage this wave's 4 rows (160 floats) into LDS, coalesced ----
    {
        v4f v = *(const v4f*)(src + lane * 4);     // 128 floats via b128
        *(v4f*)(lds + lane * 4) = v;
        lds[128 + lane] = src[128 + lane];         // remaining 32 via b32
    }
    // Same-wave produce->consume through LDS; compiler inserts the
    // s_wait_loadcnt / s_wait_dscnt dependencies. No cross-wave sharing.

    // ---- Coefficients, padded 18 -> 20 (K = 5 steps of 4) ----
    float c[20];
#pragma unroll
    for (int i = 0; i < 18; ++i) c[i] = coeff[i];
    c[18] = 0.0f; c[19] = 0.0f;

    // Per-lane A-matrix values: A[m][k] = coeff[k] for all m (rows equal, so
    // every row of D is the result vector). Lane's K slice per ISA layout:
    // lanes 0-15 -> {4s+0, 4s+1}, lanes 16-31 -> {4s+2, 4s+3}.
    float cax[5], cay[5];
#pragma unroll
    for (int s = 0; s < 5; ++s) {
        cax[s] = hi ? c[4 * s + 2] : c[4 * s + 0];
        cay[s] = hi ? c[4 * s + 3] : c[4 * s + 1];
    }

    // ---- 10 chunks of 16 outputs, processed in pairs for full-wave stores ----
#pragma unroll
    for (int p = 0; p < 5; ++p) {
        float fA[20], fB[20];
        compute_feats(lds, 32 * p + n,      fA);   // chunk 2p
        compute_feats(lds, 32 * p + 16 + n, fB);   // chunk 2p+1

        v8f accA = {};
        v8f accB = {};
#pragma unroll
        for (int s = 0; s < 5; ++s) {
            v2f a;  a.x  = cax[s];                  a.y  = cay[s];
            v2f bA; bA.x = hi ? fA[4*s+2] : fA[4*s+0];
                    bA.y = hi ? fA[4*s+3] : fA[4*s+1];
            accA = __builtin_amdgcn_wmma_f32_16x16x4_f32(
                false, a, false, bA, (short)0, accA, false, false);
            v2f bB; bB.x = hi ? fB[4*s+2] : fB[4*s+0];
                    bB.y = hi ? fB[4*s+3] : fB[4*s+1];
            accB = __builtin_amdgcn_wmma_f32_16x16x4_f32(
                false, a, false, bB, (short)0, accB, false, false);
        }

        // D layout: VGPR0, lanes 0-15 = row M=0, N=lane (all rows identical).
        // Lanes 0-15 hold chunk-A outputs, lanes 16-31 hold chunk-B outputs
        // (their duplicate half), so a single full-wave contiguous 128B store
        // covers both chunks.
        float val = hi ? accB[0] : accA[0];
        dst[32 * p + lane] = val;
    }
}

extern "C" void kernel_launch(void* const* d_in, const int* in_sizes, int n_in,
                              void* d_out, int out_size, void* d_ws, size_t ws_size,
                              hipStream_t stream) {
    // Inputs (dict order): t (int scalar, unused), u (B*40 f32), coeff (18 f32)
    const float* u     = (const float*)d_in[1];
    const float* coeff = (const float*)d_in[2];
    float* out         = (float*)d_out;

    const int total_rows = in_sizes[1] / XD;                  // 262144
    const int blocks     = (total_rows + ROWS_PER_BLOCK - 1) / ROWS_PER_BLOCK;

    l96_feat_dot_kernel<<<blocks, WAVES_PER_BLOCK * 32, 0, stream>>>(
        u, coeff, out, total_rows);
}